// SAGEProlongationGNN_64295660421654
// MI455X (gfx1250) — compile-verified
//
#include <hip/hip_runtime.h>
#include <math.h>

#define F_IN    10
#define HDIM    64
#define EDIM    3
#define NLAYERS 3

typedef __attribute__((ext_vector_type(2))) float v2f;
typedef __attribute__((ext_vector_type(8))) float v8f;

// ---------------------------------------------------------------------------
// h = relu(x @ Wp + bp)   (N,10)x(10,64) -> (N,64)
// ---------------------------------------------------------------------------
__global__ void proj_kernel(const float* __restrict__ x, const float* __restrict__ Wp,
                            const float* __restrict__ bp, float* __restrict__ h, int N) {
    int tid = blockIdx.x * blockDim.x + threadIdx.x;
    if (tid >= N * HDIM) return;
    int n = tid >> 6;
    int j = tid & 63;
    float s = bp[j];
#pragma unroll
    for (int k = 0; k < F_IN; ++k) s += x[n * F_IN + k] * Wp[k * HDIM + j];
    h[tid] = fmaxf(s, 0.f);
}

// ---------------------------------------------------------------------------
// Per-edge MLP (3 -> 16 -> 1) + sigmoid, for all 3 layers at once.
// ew1: (L,3,16)  eb1: (L,16)  ew2: (L,16,1)  eb2: (L,1)
// ---------------------------------------------------------------------------
__global__ void edge_mlp_kernel(const float* __restrict__ ea,
                                const float* __restrict__ ew1, const float* __restrict__ eb1,
                                const float* __restrict__ ew2, const float* __restrict__ eb2,
                                float* __restrict__ ew_all, int E) {
    int e = blockIdx.x * blockDim.x + threadIdx.x;
    if (e >= E) return;
    float a0 = ea[e * 3 + 0], a1 = ea[e * 3 + 1], a2 = ea[e * 3 + 2];
#pragma unroll
    for (int l = 0; l < NLAYERS; ++l) {
        const float* w1 = ew1 + l * (EDIM * 16);
        const float* b1 = eb1 + l * 16;
        const float* w2 = ew2 + l * 16;
        float s = eb2[l];
#pragma unroll
        for (int j = 0; j < 16; ++j) {
            float t = b1[j] + a0 * w1[j] + a1 * w1[16 + j] + a2 * w1[32 + j];
            s += fmaxf(t, 0.f) * w2[j];
        }
        ew_all[(size_t)l * E + e] = 1.f / (1.f + __expf(-s));
    }
}

// ---------------------------------------------------------------------------
__global__ void zero_kernel(float* __restrict__ p, long n) {
    long i = (long)blockIdx.x * blockDim.x + threadIdx.x;
    if (i < n) p[i] = 0.f;
}

// ---------------------------------------------------------------------------
// One wave32 per edge; each lane handles 2 of the 64 channels.
// agg[dst] += h[src]*w ; cnt[dst] += w  (f32 atomics, L2-resident)
// ---------------------------------------------------------------------------
__global__ void scatter_kernel(const float* __restrict__ h, const int* __restrict__ ei,
                               const float* __restrict__ ew, float* __restrict__ agg,
                               float* __restrict__ cnt, int E) {
    long tid = (long)blockIdx.x * blockDim.x + threadIdx.x;
    if (tid >= (long)E * 32) return;
    int e    = (int)(tid >> 5);
    int lane = (int)(tid & 31);
    int src = ei[e];
    int dst = ei[E + e];
    float w = ew[e];
    int c = lane * 2;
    float2 hv = *(const float2*)(h + (size_t)src * HDIM + c);
    atomicAdd(&agg[(size_t)dst * HDIM + c],     hv.x * w);
    atomicAdd(&agg[(size_t)dst * HDIM + c + 1], hv.y * w);
    if (lane == 0) atomicAdd(&cnt[dst], w);
}

// ---------------------------------------------------------------------------
__global__ void normalize_kernel(float* __restrict__ agg, const float* __restrict__ cnt, int N) {
    int i = blockIdx.x * blockDim.x + threadIdx.x;
    if (i >= N * HDIM) return;
    agg[i] = agg[i] / fmaxf(cnt[i >> 6], 1e-12f);
}

// ---------------------------------------------------------------------------
// Node update: upd = relu([h | agg] @ nw + nb) ; h = LN(h + upd)
// WMMA f32 16x16x4. Block = 256 thr = 8 waves = (2 row-tiles x 4 col-tiles)
// covering 32 nodes x 64 cols. A-frag (16x4 f32): lane m=lane&15 holds row m,
// VGPR0 = K=(lane>>4)*2, VGPR1 = K+1 (ISA 7.12.2). B-frag mirrored by K-row.
// ---------------------------------------------------------------------------
__global__ void update_wmma_kernel(float* __restrict__ h, const float* __restrict__ agg,
                                   const float* __restrict__ nw, const float* __restrict__ nb,
                                   const float* __restrict__ ln_g, const float* __restrict__ ln_b,
                                   int N) {
    __shared__ float sbuf[32 * HDIM];
    __shared__ float smu[32];
    __shared__ float srs[32];

    const int tid  = threadIdx.x;
    const int wave = tid >> 5;
    const int lane = tid & 31;
    const int rt = wave >> 2;            // row tile 0..1
    const int ct = wave & 3;             // col tile 0..3
    const int node0 = blockIdx.x * 32 + rt * 16;
    const int nidx = lane & 15;          // row (A) / col (B,C) within tile
    const int kb = (lane >> 4) * 2;      // K sub-pair select
    const int c0 = ct * 16;

    int r = node0 + nidx;
    if (r >= N) r = N - 1;               // clamp loads; stores predicated later

    v8f acc;
    {
        float cb = nb[c0 + nidx];
#pragma unroll
        for (int j = 0; j < 8; ++j) acc[j] = cb;
    }

    // K = 128 = [h(64) | agg(64)] ; kb<=2 and k%4==0 => pair never straddles 64
    for (int k = 0; k < 2 * HDIM; k += 4) {
        const float* asrc = (k < HDIM)
            ? (h   + (size_t)r * HDIM + (k + kb))
            : (agg + (size_t)r * HDIM + (k - HDIM + kb));
        v2f a;
        a.x = asrc[0];
        a.y = asrc[1];
        const float* bsrc = nw + (size_t)(k + kb) * HDIM + c0 + nidx;
        v2f b;
        b.x = bsrc[0];
        b.y = bsrc[HDIM];
        acc = __builtin_amdgcn_wmma_f32_16x16x4_f32(false, a, false, b,
                                                    (short)0, acc, false, false);
    }

    // ReLU + residual -> LDS (pre-LayerNorm values)
#pragma unroll
    for (int j = 0; j < 8; ++j) {
        int m = j + (lane >> 4) * 8;     // C/D layout: VGPRj -> M=j (lanes<16) / j+8
        int g = node0 + m;
        int gc = (g < N) ? g : (N - 1);
        float val = fmaxf(acc[j], 0.f) + h[(size_t)gc * HDIM + c0 + nidx];
        sbuf[(rt * 16 + m) * HDIM + c0 + nidx] = val;
    }
    __syncthreads();

    if (tid < 32) {
        float mu = 0.f;
        for (int c = 0; c < HDIM; ++c) mu += sbuf[tid * HDIM + c];
        mu *= (1.f / HDIM);
        float var = 0.f;
        for (int c = 0; c < HDIM; ++c) {
            float d = sbuf[tid * HDIM + c] - mu;
            var += d * d;
        }
        var *= (1.f / HDIM);
        smu[tid] = mu;
        srs[tid] = rsqrtf(var + 1e-5f);
    }
    __syncthreads();

    const int base = blockIdx.x * 32;
    for (int i = tid; i < 32 * HDIM; i += blockDim.x) {
        int ln = i >> 6;
        int c  = i & 63;
        int g  = base + ln;
        if (g < N) {
            h[(size_t)g * HDIM + c] = (sbuf[i] - smu[ln]) * srs[ln] * ln_g[c] + ln_b[c];
        }
    }
}

// ---------------------------------------------------------------------------
// out = relu(h @ hw1 + hb1) @ hw2 + hb2   (N,64)->(N,32)->(N,1)
// ---------------------------------------------------------------------------
__global__ void head_kernel(const float* __restrict__ h, const float* __restrict__ hw1,
                            const float* __restrict__ hb1, const float* __restrict__ hw2,
                            const float* __restrict__ hb2, float* __restrict__ out, int N) {
    int n = blockIdx.x * blockDim.x + threadIdx.x;
    if (n >= N) return;
    const float* hr = h + (size_t)n * HDIM;
    float o = hb2[0];
    for (int j = 0; j < 32; ++j) {
        float s = hb1[j];
        for (int k = 0; k < HDIM; ++k) s += hr[k] * hw1[k * 32 + j];
        o += fmaxf(s, 0.f) * hw2[j];
    }
    out[n] = o;
}

// ---------------------------------------------------------------------------
extern "C" void kernel_launch(void* const* d_in, const int* in_sizes, int n_in,
                              void* d_out, int out_size, void* d_ws, size_t ws_size,
                              hipStream_t stream) {
    const float* x   = (const float*)d_in[0];
    const int*   ei  = (const int*)  d_in[1];
    const float* ea  = (const float*)d_in[2];
    const float* Wp  = (const float*)d_in[3];
    const float* bp  = (const float*)d_in[4];
    const float* ew1 = (const float*)d_in[5];
    const float* eb1 = (const float*)d_in[6];
    const float* ew2 = (const float*)d_in[7];
    const float* eb2 = (const float*)d_in[8];
    const float* nw  = (const float*)d_in[9];
    const float* nb  = (const float*)d_in[10];
    const float* lng = (const float*)d_in[11];
    const float* lnb = (const float*)d_in[12];
    const float* hw1 = (const float*)d_in[13];
    const float* hb1 = (const float*)d_in[14];
    const float* hw2 = (const float*)d_in[15];
    const float* hb2 = (const float*)d_in[16];

    const int N = in_sizes[0] / F_IN;
    const int E = in_sizes[2] / EDIM;

    // Workspace layout (floats): h[N*64] | agg[N*64] | cnt[N] | ew_all[L*E]
    float* h      = (float*)d_ws;
    float* agg    = h + (size_t)N * HDIM;
    float* cnt    = agg + (size_t)N * HDIM;
    float* ew_all = cnt + N;

    const int TB = 256;

    proj_kernel<<<(N * HDIM + TB - 1) / TB, TB, 0, stream>>>(x, Wp, bp, h, N);
    edge_mlp_kernel<<<(E + TB - 1) / TB, TB, 0, stream>>>(ea, ew1, eb1, ew2, eb2, ew_all, E);

    for (int l = 0; l < NLAYERS; ++l) {
        long zn = (long)N * HDIM + N;                   // agg + cnt are contiguous
        zero_kernel<<<(int)((zn + TB - 1) / TB), TB, 0, stream>>>(agg, zn);

        long st = (long)E * 32;
        scatter_kernel<<<(int)((st + TB - 1) / TB), TB, 0, stream>>>(
            h, ei, ew_all + (size_t)l * E, agg, cnt, E);

        normalize_kernel<<<(N * HDIM + TB - 1) / TB, TB, 0, stream>>>(agg, cnt, N);

        update_wmma_kernel<<<(N + 31) / 32, TB, 0, stream>>>(
            h, agg,
            nw  + (size_t)l * 2 * HDIM * HDIM,
            nb  + l * HDIM,
            lng + l * HDIM,
            lnb + l * HDIM,
            N);
    }

    head_kernel<<<(N + TB - 1) / TB, TB, 0, stream>>>(h, hw1, hb1, hw2, hb2,
                                                      (float*)d_out, N);
}